// AttentionHead_28750511079825
// MI455X (gfx1250) — compile-verified
//
#include <hip/hip_runtime.h>
#include <math.h>

typedef __attribute__((ext_vector_type(2))) float v2f;
typedef __attribute__((ext_vector_type(4))) float v4f;
typedef __attribute__((ext_vector_type(8))) float v8f;

namespace {
constexpr int kB = 4;
constexpr int kS = 4096;
constexpr int kD = 1024;
constexpr int kH = 64;
constexpr float kLnEps = 1e-5f;
constexpr float kScale = 0.125f;  // 1/sqrt(H)
}

static __device__ __forceinline__ v8f wmma_f32(v2f a, v2f b, v8f c) {
    // D = A(16x4,f32) * B(4x16,f32) + C(16x16,f32)
    return __builtin_amdgcn_wmma_f32_16x16x4_f32(false, a, false, b, (short)0, c,
                                                 false, false);
}

// ---------------------------------------------------------------------------
// Stage 1: Q = LN(x@Wq), K = LN(x@Wk), V = x@Wv.
// 128 threads = 4 waves per 16-row tile; wave w reduces D-slice [w*256,w*256+256)
// into its own LDS slice; partials are summed, LayerNormed, stored coalesced.
// ---------------------------------------------------------------------------
__global__ __launch_bounds__(128) void proj_ln_kernel(
    const float* __restrict__ x, const float* __restrict__ Wq,
    const float* __restrict__ Wk, const float* __restrict__ Wv,
    float* __restrict__ qo, float* __restrict__ ko, float* __restrict__ vo) {
    __shared__ float tq[4][16][68];  // pitch 68: 16B-aligned rows, conflict-free
    __shared__ float tk[4][16][68];
    __shared__ float tv[4][16][68];

    const int tid = threadIdx.x;
    const int w = tid >> 5;
    const int lane = tid & 31;
    const int m = lane & 15;
    const int hi = lane >> 4;
    const int rowBase = blockIdx.x * 16;  // over B*S rows

    v8f aq[4], ak[4], av[4];
#pragma unroll
    for (int nt = 0; nt < 4; ++nt) { aq[nt] = {}; ak[nt] = {}; av[nt] = {}; }

    const float* xr = x + (size_t)(rowBase + m) * kD;
    const int k0beg = w * (kD / 4);
    // constant trip count -> scalar loop branch, EXEC stays all-1s for WMMA
    for (int it = 0; it < kD / 16; ++it) {
        const int k0 = k0beg + 4 * it;
        v2f a;
        a.x = xr[k0 + 2 * hi];
        a.y = xr[k0 + 2 * hi + 1];
        const int wrow = (k0 + 2 * hi) * kH;
#pragma unroll
        for (int nt = 0; nt < 4; ++nt) {
            const int col = nt * 16 + m;
            v2f bq, bk, bv;
            bq.x = Wq[wrow + col];      bq.y = Wq[wrow + kH + col];
            bk.x = Wk[wrow + col];      bk.y = Wk[wrow + kH + col];
            bv.x = Wv[wrow + col];      bv.y = Wv[wrow + kH + col];
            aq[nt] = wmma_f32(a, bq, aq[nt]);
            ak[nt] = wmma_f32(a, bk, ak[nt]);
            av[nt] = wmma_f32(a, bv, av[nt]);
        }
    }

    // C-layout scatter of partial sums into this wave's LDS slice
#pragma unroll
    for (int nt = 0; nt < 4; ++nt)
#pragma unroll
        for (int r = 0; r < 8; ++r) {
            const int row = r + 8 * hi;
            const int col = nt * 16 + m;
            tq[w][row][col] = aq[nt][r];
            tk[w][row][col] = ak[nt][r];
            tv[w][row][col] = av[nt][r];
        }
    __syncthreads();

    // combine 4 partial slices + LayerNorm (48 threads own one row each)
    if (tid < 48) {
        const int which = tid >> 4;  // 0=q, 1=k, 2=v
        const int row = tid & 15;
        float(*arr)[16][68] = (which == 0) ? tq : (which == 1) ? tk : tv;
        float* r0 = &arr[0][row][0];
        const float* r1 = &arr[1][row][0];
        const float* r2 = &arr[2][row][0];
        const float* r3 = &arr[3][row][0];
        float s = 0.f, s2 = 0.f;
#pragma unroll
        for (int c = 0; c < kH; ++c) {
            const float t = r0[c] + r1[c] + r2[c] + r3[c];
            r0[c] = t;
            s += t;
            s2 += t * t;
        }
        if (which < 2) {
            const float mu = s * (1.f / kH);
            const float var = s2 * (1.f / kH) - mu * mu;
            const float inv = rsqrtf(var + kLnEps);
#pragma unroll
            for (int c = 0; c < kH; ++c) r0[c] = (r0[c] - mu) * inv;
        }
    }
    __syncthreads();

    // coalesced float4 stores of the combined slice 0
    const size_t ob = (size_t)rowBase * kH;
#pragma unroll
    for (int it = 0; it < 2; ++it) {
        const int flat = it * 512 + tid * 4;
        const int row = flat >> 6;
        const int col = flat & 63;
        *reinterpret_cast<v4f*>(qo + ob + flat) =
            *reinterpret_cast<const v4f*>(&tq[0][row][col]);
        *reinterpret_cast<v4f*>(ko + ob + flat) =
            *reinterpret_cast<const v4f*>(&tk[0][row][col]);
        *reinterpret_cast<v4f*>(vo + ob + flat) =
            *reinterpret_cast<const v4f*>(&tv[0][row][col]);
    }
}

// ---------------------------------------------------------------------------
// Stage 2: flash attention, split-K over keys.
// 128 threads = 4 waves per 16-query tile; wave w handles key tiles j = w+4t.
// Row-sum of P is computed by the matrix pipe (P x ones). Partial (m,l,O) of
// the 4 waves are merged in LDS at the end.
// ---------------------------------------------------------------------------
__global__ __launch_bounds__(128) void flash_attn_kernel(
    const float* __restrict__ q, const float* __restrict__ k,
    const float* __restrict__ v, float* __restrict__ out) {
    __shared__ float ps[4][16][18];   // per-wave 16x16 P tile (8B-aligned pitch)
    __shared__ float m_s[4][16];
    __shared__ float l_s[4][16];
    __shared__ float o_s[4][16][64];  // 16 KB

    const int tid = threadIdx.x;
    const int w = tid >> 5;
    const int lane = tid & 31;
    const int m = lane & 15;
    const int hi = lane >> 4;
    const int b = blockIdx.x >> 8;  // S/16 = 256 query tiles per batch
    const int qt = blockIdx.x & 255;

    const float* Qb = q + ((size_t)b * kS + qt * 16) * kH;
    const float* Kb = k + (size_t)b * kS * kH;
    const float* Vb = v + (size_t)b * kS * kH;

    // Q A-fragments across H=64 (16 chunks of K=4), softmax scale folded in
    v2f aq[16];
#pragma unroll
    for (int kk = 0; kk < 16; ++kk) {
        const float* p = Qb + m * kH + kk * 4 + 2 * hi;
        aq[kk].x = p[0] * kScale;
        aq[kk].y = p[1] * kScale;
    }

    v8f o[4];
#pragma unroll
    for (int nt = 0; nt < 4; ++nt) o[nt] = {};
    v8f lacc = {};  // running row-sums via P x ones (all 16 cols identical)
    float mrow[8];
#pragma unroll
    for (int r = 0; r < 8; ++r) mrow[r] = -INFINITY;
    const v2f bones = {1.f, 1.f};

    constexpr int kTiles = (kS / 16) / 4;  // 64 key tiles per wave
    for (int t = 0; t < kTiles; ++t) {
        const int j = w + 4 * t;
        const float* Kt = Kb + (size_t)j * 16 * kH;

        // prefetch next tile's K/V (4 KB each = 32 lanes x 128 B) into cache
        {
            const int jn = (t + 1 < kTiles) ? (j + 4) : j;
            const size_t nb = (size_t)jn * 16 * kH + lane * 32;
            __builtin_prefetch(Kb + nb, 0, 3);
            __builtin_prefetch(Vb + nb, 0, 3);
        }

        // scores = Q*K^T : two independent 8-deep WMMA chains for ILP
        v8f sc0 = {}, sc1 = {};
#pragma unroll
        for (int kk = 0; kk < 8; ++kk) {
            v2f bk0, bk1;
            const float* p0 = Kt + m * kH + kk * 4 + 2 * hi;
            const float* p1 = p0 + 32;
            bk0.x = p0[0];  bk0.y = p0[1];
            bk1.x = p1[0];  bk1.y = p1[1];
            sc0 = wmma_f32(aq[kk], bk0, sc0);
            sc1 = wmma_f32(aq[kk + 8], bk1, sc1);
        }
        v8f sc;
#pragma unroll
        for (int r = 0; r < 8; ++r) sc[r] = sc0[r] + sc1[r];

        // online softmax: rowmax via 16-lane xor shuffles (layout-aligned)
        v8f pf;
        float al[8];
#pragma unroll
        for (int r = 0; r < 8; ++r) {
            float tmax = sc[r];
            tmax = fmaxf(tmax, __shfl_xor(tmax, 1));
            tmax = fmaxf(tmax, __shfl_xor(tmax, 2));
            tmax = fmaxf(tmax, __shfl_xor(tmax, 4));
            tmax = fmaxf(tmax, __shfl_xor(tmax, 8));
            const float mn = fmaxf(mrow[r], tmax);
            al[r] = __expf(mrow[r] - mn);
            mrow[r] = mn;
            pf[r] = __expf(sc[r] - mn);
        }
#pragma unroll
        for (int r = 0; r < 8; ++r) lacc[r] *= al[r];
#pragma unroll
        for (int nt = 0; nt < 4; ++nt)
#pragma unroll
            for (int r = 0; r < 8; ++r) o[nt][r] *= al[r];

        // transpose P (C-layout) -> A-layout via this wave's LDS tile.
        // Per-wave DS ops are in-order on CDNA5; the asm fences only stop the
        // compiler from reordering cross-lane LDS accesses.
#pragma unroll
        for (int r = 0; r < 8; ++r) ps[w][r + 8 * hi][m] = pf[r];
        asm volatile("" ::: "memory");

#pragma unroll
        for (int kk = 0; kk < 4; ++kk) {
            v2f ap;
            ap.x = ps[w][m][kk * 4 + 2 * hi];
            ap.y = ps[w][m][kk * 4 + 2 * hi + 1];
            lacc = wmma_f32(ap, bones, lacc);  // row-sums of P
            const float* vp = Vb + (size_t)(j * 16 + kk * 4 + 2 * hi) * kH;
#pragma unroll
            for (int nt = 0; nt < 4; ++nt) {
                v2f bv;
                bv.x = vp[nt * 16 + m];
                bv.y = vp[kH + nt * 16 + m];
                o[nt] = wmma_f32(ap, bv, o[nt]);
            }
        }
        asm volatile("" ::: "memory");
    }

    // publish this wave's partial state
    if (m == 0) {
#pragma unroll
        for (int r = 0; r < 8; ++r) {
            m_s[w][r + 8 * hi] = mrow[r];
            l_s[w][r + 8 * hi] = lacc[r];
        }
    }
#pragma unroll
    for (int nt = 0; nt < 4; ++nt)
#pragma unroll
        for (int r = 0; r < 8; ++r)
            o_s[w][r + 8 * hi][nt * 16 + m] = o[nt][r];
    __syncthreads();

    // merge 4 split-K partials: 8 threads per row, 8 cols per thread
    const int row = tid >> 3;
    const int c0 = (tid & 7) * 8;
    float M = m_s[0][row];
#pragma unroll
    for (int w2 = 1; w2 < 4; ++w2) M = fmaxf(M, m_s[w2][row]);
    float e[4];
    float L = 0.f;
#pragma unroll
    for (int w2 = 0; w2 < 4; ++w2) {
        e[w2] = __expf(m_s[w2][row] - M);
        L += e[w2] * l_s[w2][row];
    }
    const float invL = 1.f / L;
    float* Ob = out + ((size_t)b * kS + qt * 16) * kH;
#pragma unroll
    for (int c = 0; c < 8; ++c) {
        const float acc = e[0] * o_s[0][row][c0 + c] + e[1] * o_s[1][row][c0 + c] +
                          e[2] * o_s[2][row][c0 + c] + e[3] * o_s[3][row][c0 + c];
        Ob[row * kH + c0 + c] = acc * invL;
    }
}

extern "C" void kernel_launch(void* const* d_in, const int* in_sizes, int n_in,
                              void* d_out, int out_size, void* d_ws,
                              size_t ws_size, hipStream_t stream) {
    const float* x = (const float*)d_in[0];
    const float* Wq = (const float*)d_in[1];
    const float* Wk = (const float*)d_in[2];
    const float* Wv = (const float*)d_in[3];
    float* out = (float*)d_out;

    // workspace: q, k, v each B*S*H f32 = 4 MB (12 MB total)
    float* qws = (float*)d_ws;
    float* kws = qws + (size_t)kB * kS * kH;
    float* vws = kws + (size_t)kB * kS * kH;

    proj_ln_kernel<<<(kB * kS) / 16, 128, 0, stream>>>(x, Wq, Wk, Wv, qws, kws,
                                                       vws);
    flash_attn_kernel<<<kB * (kS / 16), 128, 0, stream>>>(qws, kws, vws, out);
}